// PhaseShifter_3100966388191
// MI455X (gfx1250) — compile-verified
//
#include <hip/hip_runtime.h>
#include <stdint.h>

typedef __attribute__((ext_vector_type(4))) float v4f;

#define THREADS 320   // 10 waves; 3200 groups/block -> exactly 5 blocks per 16000-group slice
#define ITER    10

// Fast path: BATCH=8, CUTOFF=40.
//   elements = 8*40^4 = 20,480,000  -> float4 groups = 5,120,000
//   slice (constant phase run) = 40^3 = 64,000 elems = 16,000 groups
//   block = 320 thr * 10 iter = 3,200 groups  -> 5 blocks/slice, 1600 blocks total
__global__ __launch_bounds__(THREADS)
void phase_fast(const float* __restrict__ phi,
                const float* __restrict__ sr,
                const float* __restrict__ si,
                float* __restrict__ out)
{
    const uint32_t t     = threadIdx.x;
    const uint32_t slice = blockIdx.x / 5u;            // 0..319 (SALU, const divisor)
    const uint32_t chunk = blockIdx.x - slice * 5u;    // 0..4
    const uint32_t g0    = slice * 16000u + chunk * 3200u + t;

    // [buffer][real/imag][lane] : 2*2*320*16B = 20 KB LDS
    __shared__ v4f lbuf[2][2][THREADS];

    uint32_t ldsAddr[2][2];
#pragma unroll
    for (int j = 0; j < 2; ++j) {
        ldsAddr[j][0] = (uint32_t)(uintptr_t)&lbuf[j][0][t];
        ldsAddr[j][1] = (uint32_t)(uintptr_t)&lbuf[j][1][t];
    }

    const v4f* sr4  = (const v4f*)sr;
    const v4f* si4  = (const v4f*)si;
    v4f*       out4 = (v4f*)out;

    // Prime the DMA pipeline first...
    asm volatile("global_load_async_to_lds_b128 %0, %1, off"
                 :: "v"(ldsAddr[0][0]), "v"((unsigned long long)(sr4 + g0)) : "memory");
    asm volatile("global_load_async_to_lds_b128 %0, %1, off"
                 :: "v"(ldsAddr[0][1]), "v"((unsigned long long)(si4 + g0)) : "memory");

    // ...then compute the (block-uniform) phase while the DMA is in flight.
    const uint32_t b  = slice / 40u;
    const uint32_t p  = slice - b * 40u;
    const float    th = phi[b] * (float)p;
    const float    s  = sinf(th);
    const float    c  = cosf(th);

#pragma unroll
    for (int i = 0; i < ITER; ++i) {
        if (i + 1 < ITER) {
            const uint32_t gn = g0 + (uint32_t)(i + 1) * THREADS;
            const int      jn = (i + 1) & 1;
            asm volatile("global_load_async_to_lds_b128 %0, %1, off"
                         :: "v"(ldsAddr[jn][0]), "v"((unsigned long long)(sr4 + gn)) : "memory");
            asm volatile("global_load_async_to_lds_b128 %0, %1, off"
                         :: "v"(ldsAddr[jn][1]), "v"((unsigned long long)(si4 + gn)) : "memory");
            // 2 newest in flight; the pair for iteration i is complete (in-order).
            asm volatile("s_wait_asynccnt 2" ::: "memory");
        } else {
            asm volatile("s_wait_asynccnt 0" ::: "memory");
        }

        const int      j = i & 1;
        const uint32_t g = g0 + (uint32_t)i * THREADS;

        const v4f r  = lbuf[j][0][t];   // ds_load_b128
        const v4f im = lbuf[j][1][t];   // ds_load_b128

        v4f o0, o1;                     // interleaved complex64 output
        o0.x = c * r.x - s * im.x;  o0.y = s * r.x + c * im.x;
        o0.z = c * r.y - s * im.y;  o0.w = s * r.y + c * im.y;
        o1.x = c * r.z - s * im.z;  o1.y = s * r.z + c * im.z;
        o1.z = c * r.w - s * im.w;  o1.w = s * r.w + c * im.w;

        // Write-once stream: non-temporal B128 stores keep L2 for the read stream.
        __builtin_nontemporal_store(o0, out4 + 2 * (size_t)g);
        __builtin_nontemporal_store(o1, out4 + 2 * (size_t)g + 1);
    }
}

// Generic fallback for unexpected shapes: scalar grid-stride.
__global__ void phase_generic(const float* __restrict__ phi,
                              const float* __restrict__ sr,
                              const float* __restrict__ si,
                              float* __restrict__ out,
                              uint32_t n, uint32_t slice_elems, uint32_t cut)
{
    const uint32_t idx    = blockIdx.x * blockDim.x + threadIdx.x;
    const uint32_t stride = gridDim.x * blockDim.x;
    for (uint32_t e = idx; e < n; e += stride) {
        const uint32_t slice = e / slice_elems;
        const uint32_t b     = slice / cut;
        const uint32_t p     = slice - b * cut;
        const float    th    = phi[b] * (float)p;
        const float    s     = sinf(th);
        const float    c     = cosf(th);
        const float    r     = sr[e];
        const float    im    = si[e];
        out[2 * (size_t)e]     = c * r - s * im;
        out[2 * (size_t)e + 1] = s * r + c * im;
    }
}

extern "C" void kernel_launch(void* const* d_in, const int* in_sizes, int n_in,
                              void* d_out, int out_size, void* d_ws, size_t ws_size,
                              hipStream_t stream)
{
    const float* phi = (const float*)d_in[0];   // (BATCH,)
    const float* sr  = (const float*)d_in[1];   // (BATCH, C, C, C, C)
    const float* si  = (const float*)d_in[2];
    float*       out = (float*)d_out;           // interleaved complex64

    const int batch = in_sizes[0];
    const int N     = in_sizes[1];

    if (batch == 8 && N == 20480000) {
        phase_fast<<<1600, THREADS, 0, stream>>>(phi, sr, si, out);
    } else if (N > 0 && batch > 0) {
        const long long per = (long long)N / batch;
        uint32_t cut = 1;
        for (uint32_t cc = 1; cc <= 2048; ++cc) {
            const long long c4 = (long long)cc * cc * cc * cc;
            if (c4 == per) { cut = cc; break; }
            if (c4 > per) break;
        }
        const uint32_t slice_elems = (uint32_t)(per / cut);
        uint32_t blocks = (uint32_t)((N + 255) / 256);
        if (blocks > 65535u) blocks = 65535u;
        phase_generic<<<blocks, 256, 0, stream>>>(phi, sr, si, out,
                                                  (uint32_t)N, slice_elems, cut);
    }
}